// SpMiddle2K_197568495911
// MI455X (gfx1250) — compile-verified
//
#include <hip/hip_runtime.h>

// ---------------------------------------------------------------------------
// SpMiddle2K dense implementation for MI455X (gfx1250, wave32).
// Conv layers are implicit GEMMs on V_WMMA_F32_16X16X4_F32, specialized at
// compile time on (CIN, COUT) so the K / N-tile loops fully unroll.
// One block = 4 waves = 64 output voxels; per 3x3x3 tap the block stages the
// activation rows (16 x CIN per wave, bank-conflict-free padded layout) and
// the filter slab (CIN x COUT, K-paired layout so each B fragment is one
// ds_load_b64) in LDS, then streams WMMAs from LDS fragments.
// ---------------------------------------------------------------------------

typedef float v2f __attribute__((ext_vector_type(2)));
typedef float v8f __attribute__((ext_vector_type(8)));

#define BN_EPS 1e-3f

// ----------------------------- utility kernels -----------------------------

__global__ __launch_bounds__(256) void k_zero(float* __restrict__ p, long n) {
  long i  = (long)blockIdx.x * blockDim.x + threadIdx.x;
  long st = (long)gridDim.x * blockDim.x;
  for (; i < n; i += st) p[i] = 0.f;
}

__global__ __launch_bounds__(256) void k_scatter(const float* __restrict__ vf,
                                                 const int* __restrict__ coors,
                                                 float* __restrict__ x,
                                                 float* __restrict__ mask,
                                                 int nv, int C, int H, int W) {
  long t   = (long)blockIdx.x * blockDim.x + threadIdx.x;
  long tot = (long)nv * C;
  if (t >= tot) return;
  int v = (int)(t / C);
  int c = (int)(t % C);
  int d = coors[4 * v + 1];
  int h = coors[4 * v + 2];
  int w = coors[4 * v + 3];
  long idx = ((long)d * H + h) * W + w;
  x[idx * C + c] = vf[(long)v * C + c];
  if (c == 0) mask[idx] = 1.f;
}

// mask pooling for strided ('sp') layers: new_mask = (conv(mask, ones) > 0)
__global__ __launch_bounds__(256) void k_maskpool(
    const float* __restrict__ mi, float* __restrict__ mo,
    int Di, int Hi, int Wi, int Do, int Ho, int Wo,
    int KD, int KH, int KW, int SD, int SH, int SW, int PD, int PH, int PW) {
  long t    = (long)blockIdx.x * blockDim.x + threadIdx.x;
  long nvox = (long)Do * Ho * Wo;
  if (t >= nvox) return;
  int od = (int)(t / ((long)Ho * Wo));
  int r  = (int)(t % ((long)Ho * Wo));
  int oh = r / Wo;
  int ow = r % Wo;
  float out = 0.f;
  for (int kd = 0; kd < KD; ++kd) {
    int id = od * SD - PD + kd;
    if (id < 0 || id >= Di) continue;
    for (int kh = 0; kh < KH; ++kh) {
      int ih = oh * SH - PH + kh;
      if (ih < 0 || ih >= Hi) continue;
      for (int kw = 0; kw < KW; ++kw) {
        int iw = ow * SW - PW + kw;
        if (iw < 0 || iw >= Wi) continue;
        if (mi[((long)id * Hi + ih) * Wi + iw] > 0.f) out = 1.f;
      }
    }
  }
  mo[t] = out;
}

// ------------------------- WMMA implicit-GEMM conv -------------------------
// f32 WMMA 16x16x4 fragment maps (ISA 7.12.2):
//   A: lanes l and l+16 share row M = l&15; lane half selects K pair {k,k+1}
//      (half 0) or {k+2,k+3} (half 1).
//   B: lanes 0-15 -> N = lane, rows {k,k+1}; lanes 16-31 rows {k+2,k+3}.
//   D: vgpr r -> M = r + (lane>=16 ? 8 : 0), N = lane&15.
//
// LDS layouts:
//   A rows padded to CIN+4 dwords: wave32 b64 read hits banks
//   4*row + {ka..ka+3} -> all 64 banks once, conflict-free.
//   B stored K-paired: element (ci,co) at [(ci>>1)*(2*COUT+4) + 2*co + (ci&1)]
//   so a B fragment {B[ka][n], B[ka+1][n]} is a single aligned ds_load_b64.
template <int CIN, int COUT>
__global__ __launch_bounds__(128) void k_conv_wmma(
    const float* __restrict__ x, const float* __restrict__ w,
    float* __restrict__ y,
    int Din, int Hin, int Win, int Dout, int Hout, int Wout,
    int KD, int KH, int KW, int SD, int SH, int SW, int PD, int PH, int PW) {
  constexpr int NT     = (COUT + 15) / 16;  // N tiles of 16 output channels
  constexpr int CH     = CIN / 2;           // channels staged per lane half
  constexpr int APAD   = CIN + 4;           // padded A row stride (dwords)
  constexpr int BPAIR  = 2 * COUT + 4;      // padded B pair-row stride (dwords)

  __shared__ float ldsA[4 * 16 * APAD];     // 4 waves x 16 voxel rows
  __shared__ float ldsB[(CIN / 2) * BPAIR]; // current tap's filter slab

  const int tid  = threadIdx.x;
  const int wave = tid >> 5;
  const int lane = tid & 31;
  const int half = lane >> 4;
  const int lid  = lane & 15;

  __builtin_prefetch(w, 0, 0);  // global_prefetch_b8 on the filter

  const long nvox = (long)Dout * Hout * Wout;
  const long tile = ((long)blockIdx.x * 4 + wave) * 16;

  // Output voxel for this lane's A row (lanes l and l+16 share row lid).
  long ovox   = tile + lid;
  bool mvalid = ovox < nvox;
  int od = 0, oh = 0, ow = 0;
  if (mvalid) {
    od = (int)(ovox / ((long)Hout * Wout));
    int r = (int)(ovox % ((long)Hout * Wout));
    oh = r / Wout;
    ow = r % Wout;
  }

  float* aRow = &ldsA[(wave * 16 + lid) * APAD];

  v8f acc[NT];
#pragma unroll
  for (int nt = 0; nt < NT; ++nt) acc[nt] = v8f{};

  const int ntaps = KD * KH * KW;
  for (int tap = 0; tap < ntaps; ++tap) {
    int kd = tap / (KH * KW);
    int rr = tap % (KH * KW);
    int kh = rr / KW;
    int kw = rr % KW;
    int id = od * SD - PD + kd;
    int ih = oh * SH - PH + kh;
    int iw = ow * SW - PW + kw;
    bool valid = mvalid && id >= 0 && id < Din && ih >= 0 && ih < Hin &&
                 iw >= 0 && iw < Win;
    const float* xrow =
        x + (valid ? (((long)id * Hin + ih) * Win + iw) * (long)CIN : 0);

    __syncthreads();
    // Stage A: lane stages row `lid`, channel range [half*CH, half*CH+CH).
#pragma unroll
    for (int c = 0; c < CH; c += 4) {
      int cc = half * CH + c;
      float4 vv = valid ? *(const float4*)(xrow + cc)
                        : make_float4(0.f, 0.f, 0.f, 0.f);
      *(float4*)(aRow + cc) = vv;
    }
    // Stage B: block reads the tap's contiguous CIN x COUT slab with b128
    // loads (a float4 never crosses a ci row: COUT % 4 == 0) and scatters
    // into the K-paired LDS layout.
    {
      const float* wt = w + (long)tap * (CIN * COUT);
      for (int i = tid * 4; i < CIN * COUT; i += 128 * 4) {
        float4 vv = *(const float4*)(wt + i);
        int ci  = i / COUT;
        int co0 = i % COUT;
        float* dst = &ldsB[(ci >> 1) * BPAIR + (ci & 1)];
        dst[2 * (co0 + 0)] = vv.x;
        dst[2 * (co0 + 1)] = vv.y;
        dst[2 * (co0 + 2)] = vv.z;
        dst[2 * (co0 + 3)] = vv.w;
      }
    }
    __syncthreads();

#pragma unroll
    for (int k = 0; k < CIN; k += 4) {
      const int ka = k + half * 2;       // even
      const int p  = ka >> 1;            // K pair index
      v2f a = *(const v2f*)(aRow + ka);  // one ds_load_b64
#pragma unroll
      for (int nt = 0; nt < NT; ++nt) {
        int n = nt * 16 + lid;
        v2f b;
        if (COUT % 16 == 0 || n < COUT) {
          b = *(const v2f*)(&ldsB[p * BPAIR + 2 * n]);  // one ds_load_b64
        } else {
          b.x = 0.f;
          b.y = 0.f;
        }
        acc[nt] = __builtin_amdgcn_wmma_f32_16x16x4_f32(
            /*neg_a=*/false, a, /*neg_b=*/false, b,
            /*c_mod=*/(short)0, acc[nt], /*reuse_a=*/false, /*reuse_b=*/false);
      }
    }
  }

  // Store D tiles: vgpr r holds row m = r + half*8, col n = nt*16 + (lane&15).
#pragma unroll
  for (int nt = 0; nt < NT; ++nt) {
    int n = nt * 16 + lid;
#pragma unroll
    for (int r = 0; r < 8; ++r) {
      int  m = r + half * 8;
      long v = tile + m;
      if (v < nvox && n < COUT) y[v * (long)COUT + n] = acc[nt][r];
    }
  }
}

// ------------------------------ BN statistics ------------------------------
// stats layout: [0..C) sum, [C..2C) sumsq, [2C] masked-voxel count.
__global__ __launch_bounds__(256) void k_bnstats(const float* __restrict__ y,
                                                 const float* __restrict__ mask,
                                                 long nvox, int C,
                                                 float* __restrict__ stats) {
  __shared__ float s[129];  // up to C=64: 64 sums + 64 sumsqs + 1 count
  for (int i = threadIdx.x; i < 2 * C + 1; i += blockDim.x) s[i] = 0.f;
  __syncthreads();

  long i  = (long)blockIdx.x * blockDim.x + threadIdx.x;
  long st = (long)gridDim.x * blockDim.x;
  for (long v = i; v < nvox; v += st) {
    float mk = mask[v];
    if (mk > 0.f) {
      const float* row = y + v * (long)C;
      for (int c = 0; c < C; ++c) {
        float val = row[c];
        atomicAdd(&s[c], val);
        atomicAdd(&s[C + c], val * val);
      }
      atomicAdd(&s[2 * C], 1.f);
    }
  }
  __syncthreads();
  for (int j = threadIdx.x; j < 2 * C + 1; j += blockDim.x)
    if (s[j] != 0.f) atomicAdd(&stats[j], s[j]);
}

// -------------------------- BN apply + ReLU + mask -------------------------
__global__ __launch_bounds__(256) void k_bnapply(float* __restrict__ y,
                                                 const float* __restrict__ mask,
                                                 const float* __restrict__ stats,
                                                 const float* __restrict__ g,
                                                 const float* __restrict__ bt,
                                                 long nvox, int C) {
  long i   = (long)blockIdx.x * blockDim.x + threadIdx.x;
  long tot = nvox * (long)C;
  if (i >= tot) return;
  long v = i / C;
  int  c = (int)(i % C);
  float n    = fmaxf(stats[2 * C], 1.f);
  float mean = stats[c] / n;
  float var  = stats[C + c] / n - mean * mean;
  float sc   = g[c] * rsqrtf(var + BN_EPS);
  float val  = (y[i] - mean) * sc + bt[c];
  y[i] = fmaxf(val, 0.f) * mask[v];
}

// --------------------------- final NDHWC -> (CD)HW -------------------------
__global__ __launch_bounds__(256) void k_final(const float* __restrict__ x,
                                               float* __restrict__ out,
                                               int D, int H, int W, int C) {
  int t   = blockIdx.x * blockDim.x + threadIdx.x;
  int tot = C * D * H * W;
  if (t >= tot) return;
  int w = t % W;
  int h = (t / W) % H;
  int d = (t / (W * H)) % D;
  int c = t / (W * H * D);
  out[((c * D + d) * H + h) * W + w] = x[(((long)d * H + h) * W + w) * C + c];
}

// --------------------------------- driver ----------------------------------

extern "C" void kernel_launch(void* const* d_in, const int* in_sizes, int n_in,
                              void* d_out, int out_size, void* d_ws,
                              size_t ws_size, hipStream_t stream) {
  // Input order: voxel_features, (w0,g0,b0, ..., w16,g16,b16), coors, batch
  const float* vf    = (const float*)d_in[0];
  const int*   coors = (const int*)d_in[1 + 3 * 17];
  const int    C_IN  = 128;
  const int    nv    = in_sizes[0] / C_IN;

  auto Wp = [&](int i) { return (const float*)d_in[1 + 3 * i]; };
  auto Gp = [&](int i) { return (const float*)d_in[2 + 3 * i]; };
  auto Bp = [&](int i) { return (const float*)d_in[3 + 3 * i]; };

  // Workspace layout (floats).
  const long NVOX0 = 81L * 64 * 64;       // 331776 dense grid voxels
  float* bufA  = (float*)d_ws;            // NVOX0*128 floats (~170 MB)
  float* bufB  = bufA + NVOX0 * 128;      // 4M floats (16 MB) ping buffer
  float* maskA = bufB + (4L << 20);
  float* maskB = maskA + NVOX0;
  float* stats = maskB + NVOX0;           // 129 floats
  (void)ws_size; (void)n_in; (void)out_size;

  // Scatter voxels into dense grid + build mask.
  k_zero<<<4096, 256, 0, stream>>>(bufA, NVOX0 * 128);
  k_zero<<<1024, 256, 0, stream>>>(maskA, NVOX0);
  {
    long tot = (long)nv * C_IN;
    k_scatter<<<(int)((tot + 255) / 256), 256, 0, stream>>>(
        vf, coors, bufA, maskA, nv, C_IN, 64, 64);
  }

  // Layer descriptors (kind: 0=subm, 1=sp).
  static const int kindT[17] = {0,0,1,0,0,1,0,0,1,0,0,0,1,0,0,0,1};
  static const int kT[17][3] = {
      {3,3,3},{3,3,3},{3,3,3},{3,3,3},{3,3,3},{3,3,3},{3,3,3},{3,3,3},
      {3,3,3},{3,3,3},{3,3,3},{3,3,3},{3,3,3},{3,3,3},{3,3,3},{3,3,3},{3,1,1}};
  static const int sT[17][3] = {
      {1,1,1},{1,1,1},{2,2,2},{1,1,1},{1,1,1},{2,2,2},{1,1,1},{1,1,1},
      {2,2,2},{1,1,1},{1,1,1},{1,1,1},{2,2,2},{1,1,1},{1,1,1},{1,1,1},{2,1,1}};
  static const int pT[17][3] = {
      {1,1,1},{1,1,1},{1,1,1},{1,1,1},{1,1,1},{1,1,1},{1,1,1},{1,1,1},
      {1,1,1},{1,1,1},{1,1,1},{1,1,1},{0,1,1},{1,1,1},{1,1,1},{1,1,1},{0,0,0}};
  static const int ciT[17] = {128,8,8,16,16,16,32,32,32,64,64,64,64,64,64,64,64};
  static const int coT[17] = {8,8,16,16,16,32,32,32,64,64,64,64,64,64,64,64,64};

  float* x  = bufA;
  float* y  = bufB;
  float* mc = maskA;
  float* mn = maskB;
  int D = 81, H = 64, Wd = 64;

  for (int i = 0; i < 17; ++i) {
    const int kd = kT[i][0], kh = kT[i][1], kw = kT[i][2];
    const int sd = sT[i][0], sh = sT[i][1], sw = sT[i][2];
    const int pd = pT[i][0], ph = pT[i][1], pw = pT[i][2];
    const int cin = ciT[i], cout = coT[i];

    int Do = D, Ho = H, Wo = Wd;
    if (kindT[i]) {
      Do = (D  + 2 * pd - kd) / sd + 1;
      Ho = (H  + 2 * ph - kh) / sh + 1;
      Wo = (Wd + 2 * pw - kw) / sw + 1;
      long nvo = (long)Do * Ho * Wo;
      k_maskpool<<<(int)((nvo + 255) / 256), 256, 0, stream>>>(
          mc, mn, D, H, Wd, Do, Ho, Wo, kd, kh, kw, sd, sh, sw, pd, ph, pw);
      float* t = mc; mc = mn; mn = t;
    }
    const long nvo    = (long)Do * Ho * Wo;
    const int  blocks = (int)((nvo + 63) / 64);  // 4 waves x 16 voxels each

    // Compile-time specialized implicit-GEMM conv on WMMA.
#define CONV_CASE(CI, CO)                                                     \
    if (cin == CI && cout == CO)                                              \
      k_conv_wmma<CI, CO><<<blocks, 128, 0, stream>>>(                        \
          x, Wp(i), y, D, H, Wd, Do, Ho, Wo,                                  \
          kd, kh, kw, sd, sh, sw, pd, ph, pw);
    CONV_CASE(128, 8)
    else CONV_CASE(8, 8)
    else CONV_CASE(8, 16)
    else CONV_CASE(16, 16)
    else CONV_CASE(16, 32)
    else CONV_CASE(32, 32)
    else CONV_CASE(32, 64)
    else CONV_CASE(64, 64)
#undef CONV_CASE

    // Masked BN stats + fused apply/ReLU/mask.
    k_zero<<<1, 256, 0, stream>>>(stats, 2 * cout + 1);
    long sb = (nvo + 255) / 256; if (sb > 1024) sb = 1024;
    k_bnstats<<<(int)sb, 256, 0, stream>>>(y, mc, nvo, cout, stats);
    long tot = nvo * cout;
    k_bnapply<<<(int)((tot + 255) / 256), 256, 0, stream>>>(
        y, mc, stats, Gp(i), Bp(i), nvo, cout);

    float* t = x; x = y; y = t;
    D = Do; H = Ho; Wd = Wo;
  }

  // (1, D, H, W, 64) -> (64*D, H, W) flattened.
  k_final<<<(64 * D * H * Wd + 255) / 256, 256, 0, stream>>>(
      x, (float*)d_out, D, H, Wd, 64);
}